// SS2D_17179869668
// MI455X (gfx1250) — compile-verified
//
#include <hip/hip_runtime.h>
#include <math.h>

// SS2D (VMamba selective-scan 2D) for MI455X / gfx1250, wave32.
// GEMMs -> V_WMMA_F32_16X16X4_F32 (fp32-exact), scan -> VALU with Bs/Cs staged
// through LDS via GLOBAL_LOAD_ASYNC_TO_LDS_B128 (ASYNCcnt) inline asm.

#define Bsz   8
#define Hn    64
#define Wn    64
#define Cn    96
#define Dn    192
#define Ln    4096      // H*W
#define NST   16        // D_STATE
#define DTR   6         // DT_RANK
#define KDIR  4
#define CDBL  38        // DTR + 2*NST

typedef __attribute__((ext_vector_type(2))) float v2f;
typedef __attribute__((ext_vector_type(8))) float v8f;

// Map scan-step l of direction k to canonical spatial index p = h*W + w.
// k0: hw forward, k1: wh forward, k2: hw reversed, k3: wh reversed.
// Both the u-gather from xc and the y-scatter into ysum use this same map,
// which makes the flip/transpose recombination free.
__device__ __forceinline__ int dir_spatial(int k, int l) {
  int ll = (k >= 2) ? (Ln - 1 - l) : l;
  if ((k & 1) == 0) return ll;
  int w = ll >> 6;        // ll / H
  int h = ll & 63;        // ll % H
  return (h << 6) + w;    // h*W + w
}

// Async global->LDS DMA (gfx1250). VDST operand carries the LDS byte address
// (low 32 bits of the generic pointer per ISA flat-aperture rules); tracked by
// ASYNCcnt, drained with s_wait_asynccnt.
__device__ __forceinline__ void async_g2l_b128(void* lds_dst, const void* gsrc) {
  unsigned lds_off = (unsigned)(size_t)lds_dst;
  asm volatile("global_load_async_to_lds_b128 %0, %1, off"
               :: "v"(lds_off), "v"(gsrc) : "memory");
}
__device__ __forceinline__ void async_wait0() {
  asm volatile("s_wait_asynccnt 0" ::: "memory");
}

// ---------------- in_proj GEMM: xz = x(BL x 96) @ W^T(96 x 384) -------------
// One wave per 16x16 output tile. K = 96 -> 24 WMMA f32 16x16x4 steps.
__global__ __launch_bounds__(32)
void k_inproj(const float* __restrict__ x, const float* __restrict__ w,
              float* __restrict__ xin, float* __restrict__ zbuf) {
  const int m0   = blockIdx.x * 16;
  const int n0   = blockIdx.y * 16;
  const int ln   = threadIdx.x;
  const int half = (ln < 16) ? 0 : 2;
  const int lidx = ln & 15;
  const float* xr = x + (size_t)(m0 + lidx) * Cn;   // A row (M = lidx)
  const float* wr = w + (size_t)(n0 + lidx) * Cn;   // B col (N = lidx), W row-major (384,96)
  v8f acc = {0.f,0.f,0.f,0.f,0.f,0.f,0.f,0.f};
  #pragma unroll 4
  for (int kk = 0; kk < Cn; kk += 4) {
    v2f a, b;
    a.x = xr[kk + half];     a.y = xr[kk + half + 1];
    b.x = wr[kk + half];     b.y = wr[kk + half + 1];
    acc = __builtin_amdgcn_wmma_f32_16x16x4_f32(false, a, false, b, (short)0, acc,
                                                false, false);
  }
  // n0 is 16-aligned and the xin/z split is at 192 (16-aligned) -> uniform.
  float* dst;
  int col;
  if (n0 < Dn) { dst = xin;  col = n0 + lidx; }
  else         { dst = zbuf; col = n0 - Dn + lidx; }
  const int rbase = m0 + ((ln < 16) ? 0 : 8);
  #pragma unroll
  for (int r = 0; r < 8; ++r)
    dst[(size_t)(rbase + r) * Dn + col] = acc[r];
}

// ---------------- depthwise 3x3 conv + bias + SiLU --------------------------
// xin (B,L,D) channel-inner  ->  xc (B,D,H,W) spatial-inner (scan layout).
__global__ void k_dwconv(const float* __restrict__ xin, const float* __restrict__ cw,
                         const float* __restrict__ cb, float* __restrict__ xc) {
  const int idx = blockIdx.x * blockDim.x + threadIdx.x;
  if (idx >= Bsz * Dn * Ln) return;
  const int sp = idx & (Ln - 1);
  const int d  = (idx >> 12) % Dn;
  const int b  = idx / (Dn * Ln);
  const int h = sp >> 6, w = sp & 63;
  float acc = cb[d];
  #pragma unroll
  for (int dy = 0; dy < 3; ++dy) {
    const int hh = h + dy - 1;
    if (hh < 0 || hh >= Hn) continue;
    #pragma unroll
    for (int dx = 0; dx < 3; ++dx) {
      const int ww = w + dx - 1;
      if (ww < 0 || ww >= Wn) continue;
      acc += cw[d * 9 + dy * 3 + dx] *
             xin[((size_t)b * Ln + hh * Wn + ww) * Dn + d];
    }
  }
  xc[((size_t)b * Dn + d) * Ln + sp] = acc / (1.f + __expf(-acc));   // SiLU
}

// ---------------- x_dbl GEMM per direction: (38x192) @ xs(192 x L) ----------
// B operand columns are gathered straight from xc via dir_spatial (no xs buffer).
__device__ __forceinline__ void store_dbl(float* __restrict__ dts, float* __restrict__ Bsb,
                                          float* __restrict__ Csb, int b, int row, int l,
                                          float v) {
  if (row < DTR)
    dts[((size_t)b * DTR + row) * Ln + l] = v;
  else if (row < DTR + NST)
    Bsb[((size_t)b * NST + (row - DTR)) * Ln + l] = v;
  else if (row < CDBL)
    Csb[((size_t)b * NST + (row - DTR - NST)) * Ln + l] = v;
}

__global__ __launch_bounds__(32)
void k_xdbl(const float* __restrict__ xc, const float* __restrict__ xw,
            float* __restrict__ dts, float* __restrict__ Bsb,
            float* __restrict__ Csb, int k) {
  const int b    = blockIdx.x >> 8;           // 256 l-tiles per batch
  const int l0   = (blockIdx.x & 255) << 4;
  const int ln   = threadIdx.x;
  const int half = (ln < 16) ? 0 : 2;
  const int lidx = ln & 15;
  const int l    = l0 + lidx;
  const int sp   = dir_spatial(k, l);
  const float* colp = xc + (size_t)b * Dn * Ln + sp;    // column of xs: stride Ln per d
  const float* Wp   = xw + k * CDBL * Dn;
  const int r0 = lidx, r1 = 16 + lidx, r2 = 32 + lidx;  // A rows for the 3 M-tiles
  v8f acc0 = {0.f,0.f,0.f,0.f,0.f,0.f,0.f,0.f};
  v8f acc1 = acc0, acc2 = acc0;
  for (int kk = 0; kk < Dn; kk += 4) {
    v2f bf, a0, a1, a2;
    bf.x = colp[(size_t)(kk + half) * Ln];
    bf.y = colp[(size_t)(kk + half + 1) * Ln];
    a0.x = Wp[r0 * Dn + kk + half];  a0.y = Wp[r0 * Dn + kk + half + 1];
    a1.x = Wp[r1 * Dn + kk + half];  a1.y = Wp[r1 * Dn + kk + half + 1];
    a2.x = (r2 < CDBL) ? Wp[r2 * Dn + kk + half]     : 0.f;
    a2.y = (r2 < CDBL) ? Wp[r2 * Dn + kk + half + 1] : 0.f;
    acc0 = __builtin_amdgcn_wmma_f32_16x16x4_f32(false, a0, false, bf, (short)0, acc0, false, false);
    acc1 = __builtin_amdgcn_wmma_f32_16x16x4_f32(false, a1, false, bf, (short)0, acc1, false, false);
    acc2 = __builtin_amdgcn_wmma_f32_16x16x4_f32(false, a2, false, bf, (short)0, acc2, false, false);
  }
  const int rb = (ln < 16) ? 0 : 8;
  #pragma unroll
  for (int r = 0; r < 8; ++r) {
    store_dbl(dts, Bsb, Csb, b,      r + rb, l, acc0[r]);
    store_dbl(dts, Bsb, Csb, b, 16 + r + rb, l, acc1[r]);
    store_dbl(dts, Bsb, Csb, b, 32 + r + rb, l, acc2[r]);
  }
}

// ---------------- delta = softplus( Wdt(192x6) @ dts + bias ) ---------------
// K padded 6 -> 8 (two 16x16x4 steps, zero-fill r>=6). Output (B,L,D).
__global__ __launch_bounds__(32)
void k_delta(const float* __restrict__ dts, const float* __restrict__ wd,
             const float* __restrict__ bias, float* __restrict__ delta, int k) {
  const int m0   = blockIdx.x * 16;
  const int b    = blockIdx.y >> 8;
  const int l0   = (blockIdx.y & 255) << 4;
  const int ln   = threadIdx.x;
  const int half = (ln < 16) ? 0 : 2;
  const int lidx = ln & 15;
  const int l    = l0 + lidx;
  const float* Wd = wd + k * Dn * DTR;
  const int arow = m0 + lidx;
  v8f acc = {0.f,0.f,0.f,0.f,0.f,0.f,0.f,0.f};
  #pragma unroll
  for (int kk = 0; kk < 8; kk += 4) {
    const int q0 = kk + half, q1 = kk + half + 1;
    v2f a, bf;
    a.x  = (q0 < DTR) ? Wd[arow * DTR + q0] : 0.f;
    a.y  = (q1 < DTR) ? Wd[arow * DTR + q1] : 0.f;
    bf.x = (q0 < DTR) ? dts[((size_t)b * DTR + q0) * Ln + l] : 0.f;
    bf.y = (q1 < DTR) ? dts[((size_t)b * DTR + q1) * Ln + l] : 0.f;
    acc = __builtin_amdgcn_wmma_f32_16x16x4_f32(false, a, false, bf, (short)0, acc, false, false);
  }
  const int rb = m0 + ((ln < 16) ? 0 : 8);
  const float* bs = bias + k * Dn;
  #pragma unroll
  for (int r = 0; r < 8; ++r) {
    const int row = rb + r;
    float v = acc[r] + bs[row];
    v = (v > 20.f) ? v : log1pf(__expf(v));            // softplus
    delta[((size_t)b * Ln + l) * Dn + row] = v;
  }
}

// ---------------- selective scan (sequential in L), one direction -----------
// 192 lanes = one channel each; Bs/Cs staged through LDS in 64-step chunks via
// async global->LDS b128 DMA (no VGPR round-trip), drained on ASYNCcnt.
#define CHUNK 64
__global__ __launch_bounds__(Dn)
void k_scan(const float* __restrict__ xc, const float* __restrict__ delta,
            const float* __restrict__ Bsb, const float* __restrict__ Csb,
            const float* __restrict__ A_logs, const float* __restrict__ Ds,
            float* __restrict__ ysum, int k, int init) {
  __shared__ __align__(16) float sB[NST * CHUNK];
  __shared__ __align__(16) float sC[NST * CHUNK];
  const int b = blockIdx.x;
  const int d = threadIdx.x;
  float Ad[NST];
  #pragma unroll
  for (int n = 0; n < NST; ++n)
    Ad[n] = -__expf(A_logs[(size_t)(k * Dn + d) * NST + n]);
  const float Dval = Ds[k * Dn + d];
  const float* up = xc + ((size_t)b * Dn + d) * Ln;
  float h[NST];
  #pragma unroll
  for (int n = 0; n < NST; ++n) h[n] = 0.f;

  for (int lc = 0; lc < Ln; lc += CHUNK) {
    __syncthreads();                       // previous chunk fully consumed
    // 2*16 rows x (64 floats) = 512 x 16B async DMAs across 192 lanes.
    for (int t = d; t < 2 * NST * (CHUNK / 4); t += Dn) {
      const int cs = t >> 8;               // 0 -> B, 1 -> C
      const int idx = t & 255;
      const int n  = idx >> 4;             // state row
      const int j4 = (idx & 15) << 2;      // float offset within row
      const float* g = (cs ? Csb : Bsb) + ((size_t)b * NST + n) * Ln + lc + j4;
      float* s = (cs ? sC : sB) + (n << 6) + j4;
      async_g2l_b128(s, g);
    }
    async_wait0();                         // drain this wave's ASYNCcnt
    __syncthreads();                       // make LDS visible to all waves
    if (lc + CHUNK < Ln)                   // stream-ahead hint for delta
      __builtin_prefetch(&delta[((size_t)b * Ln + lc + CHUNK) * Dn + d], 0, 1);
    for (int j = 0; j < CHUNK; ++j) {
      const int l  = lc + j;
      const int sp = dir_spatial(k, l);
      const float u  = up[sp];
      const float dl = delta[((size_t)b * Ln + l) * Dn + d];
      const float du = dl * u;
      float y = Dval * u;
      #pragma unroll
      for (int n = 0; n < NST; ++n) {
        const float hn = h[n] * __expf(dl * Ad[n]) + du * sB[(n << 6) + j];
        h[n] = hn;
        y += hn * sC[(n << 6) + j];
      }
      float* yp = ysum + ((size_t)b * Ln + sp) * Dn + d;  // canonical pixel
      if (init) *yp = y; else *yp += y;
    }
  }
}

// ---------------- layernorm(D) * SiLU(z) ------------------------------------
__global__ __launch_bounds__(Dn)
void k_lngate(const float* __restrict__ ysum, const float* __restrict__ zbuf,
              const float* __restrict__ gamma, const float* __restrict__ beta,
              float* __restrict__ yg) {
  __shared__ float s1[Dn], s2[Dn];
  const int pix = blockIdx.x;
  const int d = threadIdx.x;
  const float y = ysum[(size_t)pix * Dn + d];
  s1[d] = y; s2[d] = y * y;
  __syncthreads();
  for (int off = 96; off >= 3; off >>= 1) {              // 192 = 3*2^6
    if (d < off) { s1[d] += s1[d + off]; s2[d] += s2[d + off]; }
    __syncthreads();
  }
  const float inv = 1.f / (float)Dn;
  const float mu  = (s1[0] + s1[1] + s1[2]) * inv;
  const float ms  = (s2[0] + s2[1] + s2[2]) * inv;
  const float rstd = rsqrtf(ms - mu * mu + 1e-5f);
  const float yn = (y - mu) * rstd * gamma[d] + beta[d];
  const float zv = zbuf[(size_t)pix * Dn + d];
  yg[(size_t)pix * Dn + d] = yn * (zv / (1.f + __expf(-zv)));
}

// ---------------- out_proj GEMM: out = yg(BL x 192) @ W^T(192 x 96) ---------
__global__ __launch_bounds__(32)
void k_outproj(const float* __restrict__ yg, const float* __restrict__ w,
               float* __restrict__ out) {
  const int m0   = blockIdx.x * 16;
  const int n0   = blockIdx.y * 16;
  const int ln   = threadIdx.x;
  const int half = (ln < 16) ? 0 : 2;
  const int lidx = ln & 15;
  const float* yr = yg + (size_t)(m0 + lidx) * Dn;
  const float* wr = w  + (size_t)(n0 + lidx) * Dn;      // out_proj_w (96,192)
  v8f acc = {0.f,0.f,0.f,0.f,0.f,0.f,0.f,0.f};
  #pragma unroll 4
  for (int kk = 0; kk < Dn; kk += 4) {
    v2f a, bf;
    a.x  = yr[kk + half];  a.y  = yr[kk + half + 1];
    bf.x = wr[kk + half];  bf.y = wr[kk + half + 1];
    acc = __builtin_amdgcn_wmma_f32_16x16x4_f32(false, a, false, bf, (short)0, acc, false, false);
  }
  const int rbase = m0 + ((ln < 16) ? 0 : 8);
  #pragma unroll
  for (int r = 0; r < 8; ++r)
    out[(size_t)(rbase + r) * Cn + n0 + lidx] = acc[r];
}

extern "C" void kernel_launch(void* const* d_in, const int* in_sizes, int n_in,
                              void* d_out, int out_size, void* d_ws, size_t ws_size,
                              hipStream_t stream) {
  (void)in_sizes; (void)n_in; (void)out_size; (void)ws_size;
  const float* x      = (const float*)d_in[0];
  const float* w_in   = (const float*)d_in[1];
  const float* conv_w = (const float*)d_in[2];
  const float* conv_b = (const float*)d_in[3];
  const float* x_proj = (const float*)d_in[4];
  const float* dt_w   = (const float*)d_in[5];
  const float* dt_b   = (const float*)d_in[6];
  const float* A_logs = (const float*)d_in[7];
  const float* Ds     = (const float*)d_in[8];
  const float* gamma  = (const float*)d_in[9];
  const float* beta   = (const float*)d_in[10];
  const float* w_out  = (const float*)d_in[11];

  float* ws = (float*)d_ws;
  const size_t BLD = (size_t)Bsz * Ln * Dn;            // 6,291,456 floats
  float* xin   = ws;                                   // (B,L,D)
  float* zbuf  = xin  + BLD;                           // (B,L,D)
  float* xc    = zbuf + BLD;                           // (B,D,L)
  float* ysum  = xc   + BLD;                           // (B,L,D) canonical
  float* delta = ysum + BLD;                           // (B,L,D), reused as yg
  float* dts   = delta + BLD;                          // (B,6,L)  per-direction
  float* Bsb   = dts  + (size_t)Bsz * DTR * Ln;        // (B,16,L) per-direction
  float* Csb   = Bsb  + (size_t)Bsz * NST * Ln;        // (B,16,L) per-direction

  dim3 g1((Bsz * Ln) / 16, (2 * Dn) / 16);             // 2048 x 24 tiles
  k_inproj<<<g1, 32, 0, stream>>>(x, w_in, xin, zbuf);

  k_dwconv<<<(Bsz * Dn * Ln + 255) / 256, 256, 0, stream>>>(xin, conv_w, conv_b, xc);

  for (int k = 0; k < KDIR; ++k) {                     // directions serialized
    k_xdbl<<<Bsz * (Ln / 16), 32, 0, stream>>>(xc, x_proj, dts, Bsb, Csb, k);
    dim3 gd(Dn / 16, Bsz * (Ln / 16));
    k_delta<<<gd, 32, 0, stream>>>(dts, dt_w, dt_b, delta, k);
    k_scan<<<Bsz, Dn, 0, stream>>>(xc, delta, Bsb, Csb, A_logs, Ds, ysum, k,
                                   (k == 0) ? 1 : 0);
  }

  k_lngate<<<Bsz * Ln, Dn, 0, stream>>>(ysum, zbuf, gamma, beta, delta);

  dim3 g2((Bsz * Ln) / 16, Cn / 16);
  k_outproj<<<g2, 32, 0, stream>>>(delta, w_out, (float*)d_out);
}